// RenderingKaolin_60215441490033
// MI455X (gfx1250) — compile-verified
//
#include <hip/hip_runtime.h>
#include <math.h>

typedef float v2f __attribute__((ext_vector_type(2)));
typedef float v8f __attribute__((ext_vector_type(8)));

namespace {
constexpr int kNF = 2, kT = 8, kNR = 16;          // renders = kNF*kT
constexpr int kNV = 514, kNFACE = 1024;
constexpr int kH = 128, kW = 128, kHW = kH * kW;
constexpr int kTEXH = 256, kTEXW = 256;
constexpr int kGroupFaces = 128;                  // faces staged per LDS group
constexpr int kNGroups = kNFACE / kGroupFaces;    // 8

// workspace layout (float units)
constexpr size_t WS_RM    = 0;                               // 16 * 12 (9 used)
constexpr size_t WS_TRANS = WS_RM + (size_t)kNR * 12;        // 16 * 4
constexpr size_t WS_COEF  = WS_TRANS + (size_t)kNR * 4;      // 16 * 1024 * 16
constexpr size_t WS_RGB   = WS_COEF + (size_t)kNR * kNFACE * 16;
constexpr size_t WS_MASK  = WS_RGB + (size_t)kNR * 3 * kHW;
constexpr size_t WS_ME    = WS_MASK + (size_t)kNR * kHW;
constexpr size_t WS_T0    = WS_ME + (size_t)kNR * kHW;       // 16*3*HW
constexpr size_t WS_T1    = WS_T0 + (size_t)kNR * 3 * kHW;   // 16*HW
} // namespace

// normalized 11-tap Gaussian, sigma = 1
__device__ __constant__ float GW[11] = {
    1.4867195e-06f, 1.3383022e-04f, 4.4318484e-03f, 5.3990967e-02f,
    2.4197073e-01f, 3.9894228e-01f, 2.4197073e-01f, 5.3990967e-02f,
    4.4318484e-03f, 1.3383022e-04f, 1.4867195e-06f};

// ---------------- rotation / translation prep ----------------
__device__ inline void quat_to_aa(const float q[4], float aa[3]) {
  float x = q[1], y = q[2], z = q[3];
  float sin2 = x * x + y * y + z * z;
  float sint = sqrtf(fmaxf(sin2, 1e-12f));
  float cost = q[0];
  float two_theta =
      2.0f * ((cost < 0.0f) ? atan2f(-sint, -cost) : atan2f(sint, cost));
  float k = (sin2 > 1e-12f) ? (two_theta / sint) : 2.0f;
  aa[0] = k * x; aa[1] = k * y; aa[2] = k * z;
}

__device__ inline void aa_to_R(const float aa[3], float R[9]) {
  float th = sqrtf(aa[0] * aa[0] + aa[1] * aa[1] + aa[2] * aa[2]);
  float inv = 1.0f / fmaxf(th, 1e-8f);
  float x = aa[0] * inv, y = aa[1] * inv, z = aa[2] * inv;
  float K[9] = {0.f, -z, y, z, 0.f, -x, -y, x, 0.f};
  float K2[9];
  for (int i = 0; i < 3; ++i)
    for (int j = 0; j < 3; ++j) {
      float s = 0.f;
      for (int m = 0; m < 3; ++m) s += K[i * 3 + m] * K[m * 3 + j];
      K2[i * 3 + j] = s;
    }
  float s = sinf(th), c = cosf(th);
  for (int i = 0; i < 9; ++i) R[i] = s * K[i] + (1.0f - c) * K2[i];
  R[0] += 1.0f; R[4] += 1.0f; R[8] += 1.0f;
}

__global__ void prep_rt_kernel(const float* __restrict__ quat,
                               const float* __restrict__ trans,
                               float* __restrict__ ws) {
  int f = threadIdx.x;
  if (f >= kNF) return;
  float q0[4], q1[4];
  for (int k = 0; k < 4; ++k) {
    q0[k] = quat[((f * 2) + 0) * 4 + k];
    q1[k] = quat[((f * 2) + 1) * 4 + k];
  }
  float aa[3], R0[9], Rs[9];
  quat_to_aa(q1, aa);
  aa_to_R(aa, R0);
  quat_to_aa(q0, aa);
  for (int i = 0; i < 3; ++i) aa[i] *= (1.0f / (float)kT); // /T/ROT_DIV
  aa_to_R(aa, Rs);
  float ts[3], td[3];
  for (int i = 0; i < 3; ++i) {
    ts[i] = trans[(i * kNF + f) * 2 + 1];
    td[i] = trans[(i * kNF + f) * 2 + 0];
  }
  float Rm[9];
  for (int i = 0; i < 9; ++i) Rm[i] = R0[i];
  for (int t = 0; t < kT; ++t) {
    int r = f * kT + t;
    for (int i = 0; i < 9; ++i) ws[WS_RM + r * 12 + i] = Rm[i];
    float ti = (float)t / (float)(kT - 1);
    for (int i = 0; i < 3; ++i)
      ws[WS_TRANS + r * 4 + i] = ts[i] + ti * td[i];
    float Rn[9];
    for (int i = 0; i < 3; ++i)
      for (int j = 0; j < 3; ++j) {
        float s = 0.f;
        for (int m = 0; m < 3; ++m) s += Rm[i * 3 + m] * Rs[m * 3 + j];
        Rn[i * 3 + j] = s;
      }
    for (int i = 0; i < 9; ++i) Rm[i] = Rn[i];
  }
}

// ---------------- per-face affine coefficient table ----------------
// Per face, 16 floats: rows q=0..3 = (bary0,bary1,bary2,z) affine coeffs
// [cx, cy, c1, 0], premultiplied by clamped 1/area. Invalid (nz<=0) faces
// get row0 = (0,0,-1,0) so the inside test always fails.
__global__ void prep_faces_kernel(const float* __restrict__ verts,
                                  const int* __restrict__ faces,
                                  float* __restrict__ ws) {
  __shared__ float vs[kNV * 3];
  int r = blockIdx.x;
  const float* Rm = ws + WS_RM + r * 12;
  const float* tr = ws + WS_TRANS + r * 4;
  for (int n = threadIdx.x; n < kNV; n += blockDim.x) {
    float x = verts[n * 3 + 0], y = verts[n * 3 + 1], z = verts[n * 3 + 2];
    vs[n * 3 + 0] = Rm[0] * x + Rm[1] * y + Rm[2] * z + tr[0];
    vs[n * 3 + 1] = Rm[3] * x + Rm[4] * y + Rm[5] * z + tr[1];
    vs[n * 3 + 2] = Rm[6] * x + Rm[7] * y + Rm[8] * z + tr[2] - 2.0f; // CAM_D
  }
  __syncthreads();
  const float pscale = 1.0f / tanf(0.3925f); // 1/tan(FOV/2), W==H
  for (int fi = threadIdx.x; fi < kNFACE; fi += blockDim.x) {
    int i0 = faces[fi * 3 + 0], i1 = faces[fi * 3 + 1], i2 = faces[fi * 3 + 2];
    float p0x = vs[i0 * 3], p0y = vs[i0 * 3 + 1], p0z = vs[i0 * 3 + 2];
    float p1x = vs[i1 * 3], p1y = vs[i1 * 3 + 1], p1z = vs[i1 * 3 + 2];
    float p2x = vs[i2 * 3], p2y = vs[i2 * 3 + 1], p2z = vs[i2 * 3 + 2];
    float s0x = p0x * pscale / (-p0z), s0y = p0y * pscale / (-p0z);
    float s1x = p1x * pscale / (-p1z), s1y = p1y * pscale / (-p1z);
    float s2x = p2x * pscale / (-p2z), s2y = p2y * pscale / (-p2z);
    // face normal z (camera space)
    float e1x = p1x - p0x, e1y = p1y - p0y, e1z = p1z - p0z;
    float e2x = p2x - p0x, e2y = p2y - p0y, e2z = p2z - p0z;
    float nx = e1y * e2z - e1z * e2y;
    float ny = e1z * e2x - e1x * e2z;
    float nzc = e1x * e2y - e1y * e2x;
    float nzn = nzc / (sqrtf(nx * nx + ny * ny + nzc * nzc) + 1e-8f);
    float area = (s1x - s0x) * (s2y - s0y) - (s1y - s0y) * (s2x - s0x);
    if (fabsf(area) < 1e-10f) area = 1e-10f;
    float ia = 1.0f / area;
    // edge(va,vb): w = -dy*X + dx*Y + (dy*ax - dx*ay)
    float rr[3][3];
    {
      float dx = s2x - s1x, dy = s2y - s1y; // edge(v1,v2)
      rr[0][0] = -dy; rr[0][1] = dx; rr[0][2] = dy * s1x - dx * s1y;
    }
    {
      float dx = s0x - s2x, dy = s0y - s2y; // edge(v2,v0)
      rr[1][0] = -dy; rr[1][1] = dx; rr[1][2] = dy * s2x - dx * s2y;
    }
    {
      float dx = s1x - s0x, dy = s1y - s0y; // edge(v0,v1)
      rr[2][0] = -dy; rr[2][1] = dx; rr[2][2] = dy * s0x - dx * s0y;
    }
    float c[16];
    for (int k = 0; k < 3; ++k) {
      c[0 * 4 + k] = rr[0][k] * ia;
      c[1 * 4 + k] = rr[1][k] * ia;
      c[2 * 4 + k] = rr[2][k] * ia;
      c[3 * 4 + k] = (p0z * rr[0][k] + p1z * rr[1][k] + p2z * rr[2][k]) * ia;
    }
    c[3] = c[7] = c[11] = c[15] = 0.0f;
    if (!(nzn > 0.0f)) { c[0] = 0.0f; c[1] = 0.0f; c[2] = -1.0f; }
    float4* dst = (float4*)(ws + WS_COEF + ((size_t)r * kNFACE + fi) * 16);
    dst[0] = make_float4(c[0], c[1], c[2], c[3]);
    dst[1] = make_float4(c[4], c[5], c[6], c[7]);
    dst[2] = make_float4(c[8], c[9], c[10], c[11]);
    dst[3] = make_float4(c[12], c[13], c[14], c[15]);
  }
}

// ---------------- WMMA rasterizer ----------------
// Block = 256 threads = 8 waves = one image row (8 strips of 16 pixels).
// The block stages 128-face groups of the coefficient table into LDS with a
// register-buffered pipeline (global loads for group g+1 issued before the
// compute over group g). Per 4-face chunk:
//   D(16x16) = A(16x4: 4 faces x [b0,b1,b2,z] coeff rows) x B(4x16: [X,Y,1,0])
// D layout (wave32, f32): lane n<16 -> pixel n rows 0..7 (faces 0,1),
// lane n+16 -> pixel n rows 8..15 (faces 2,3): z-argmax is lane-local
// (branchless cndmask selects), one shfl_xor(16) merges halves at the end.
__global__ __launch_bounds__(256) void raster_kernel(
    const float* __restrict__ ws_ro, float* __restrict__ ws,
    const float* __restrict__ face_feats, const float* __restrict__ tex) {
  __shared__ float4 sbuf[kGroupFaces * 4];  // 128 faces * 16 floats = 8KB

  const int r = blockIdx.y;
  const int y = blockIdx.x;
  const int tid = threadIdx.x;
  const int lane = tid & 31;
  const int xg = (tid >> 5) * 16;  // wave id * 16 -> pixel strip
  const int pix = lane & 15;

  const float X = -1.0f + 2.0f * (float)(xg + pix) / (float)(kW - 1);
  const float Y = 1.0f - 2.0f * (float)y / (float)(kH - 1);

  // B fragment: lane n<16 holds K=0,1 = (X,Y); lane n+16 holds K=2,3 = (1,0)
  v2f bfrag;
  bfrag.x = (lane < 16) ? X : 1.0f;
  bfrag.y = (lane < 16) ? Y : 0.0f;

  // A fragment addressing: lane L holds A[M=L&15][K = 2*(L>>4) + {0,1}]
  const int m = lane & 15;
  const int khalf = lane >> 4;
  const int q = m & 3;
  const int fic = m >> 2;
  const int aoff = fic * 8 + q * 2 + khalf;  // float2 offset within a 4-face chunk

  const float4* gco4 =
      (const float4*)(ws_ro + WS_COEF + (size_t)r * kNFACE * 16);
  const float2* sf2 = (const float2*)&sbuf[0];

  float bz = -INFINITY;
  int bf = -1;
  float bb0 = 0.f, bb1 = 0.f, bb2 = 0.f;

  // prime the register pipeline with group 0 (512 float4 per group)
  float4 r0 = gco4[tid];
  float4 r1 = gco4[tid + 256];

  for (int g = 0; g < kNGroups; ++g) {
    sbuf[tid] = r0;
    sbuf[tid + 256] = r1;
    __syncthreads();
    if (g + 1 < kNGroups) {
      const float4* nb = gco4 + (size_t)(g + 1) * 512;
      __builtin_prefetch(nb, 0, 1);  // global_prefetch_b8 -> warm next group
      r0 = nb[tid];
      r1 = nb[tid + 256];
    }
    // 32 chunks of 4 faces, A-fragment prefetched one chunk ahead from LDS
    float2 acur = sf2[aoff];
#pragma unroll 4
    for (int cg = 0; cg < kGroupFaces / 4; ++cg) {
      float2 anext = sf2[((cg + 1) & 31) * 32 + aoff];
      v2f afrag; afrag.x = acur.x; afrag.y = acur.y;
      v8f cz = {};
      v8f d = __builtin_amdgcn_wmma_f32_16x16x4_f32(
          false, afrag, false, bfrag, (short)0, cz, false, false);
      const int fA = g * kGroupFaces + cg * 4 + khalf * 2;
      // branchless argmax updates (first-index tie-break via strict >)
      {
        float w0 = d[0], w1 = d[1], w2 = d[2], z = d[3];
        bool t = (w0 >= 0.f) & (w1 >= 0.f) & (w2 >= 0.f) & (z > bz);
        bz = t ? z : bz;
        bf = t ? fA : bf;
        bb0 = t ? w0 : bb0;
        bb1 = t ? w1 : bb1;
        bb2 = t ? w2 : bb2;
      }
      {
        float w0 = d[4], w1 = d[5], w2 = d[6], z = d[7];
        bool t = (w0 >= 0.f) & (w1 >= 0.f) & (w2 >= 0.f) & (z > bz);
        bz = t ? z : bz;
        bf = t ? (fA + 1) : bf;
        bb0 = t ? w0 : bb0;
        bb1 = t ? w1 : bb1;
        bb2 = t ? w2 : bb2;
      }
      acur = anext;
    }
    __syncthreads();
  }

  // merge the two lane halves (argmax, first-index tie-break)
  float oz = __shfl_xor(bz, 16, 32);
  int of = __shfl_xor(bf, 16, 32);
  float o0 = __shfl_xor(bb0, 16, 32);
  float o1 = __shfl_xor(bb1, 16, 32);
  float o2 = __shfl_xor(bb2, 16, 32);
  bool take = (of >= 0) && ((bf < 0) || (oz > bz) || (oz == bz && of < bf));
  if (take) { bz = oz; bf = of; bb0 = o0; bb1 = o1; bb2 = o2; }

  if (lane < 16) {
    const int px = xg + pix;
    float hit = (bf >= 0) ? 1.0f : 0.0f;
    int sf = (bf >= 0) ? bf : 0;
    const float* ff = face_feats + (size_t)sf * 6;
    float u = (bb0 * ff[0] + bb1 * ff[2] + bb2 * ff[4]) * hit;
    float v = (bb0 * ff[1] + bb1 * ff[3] + bb2 * ff[5]) * hit;
    // bilinear texture sample (matches reference clamp/flip semantics)
    float xs = u * (float)(kTEXW - 1);
    float ysv = (1.0f - v) * (float)(kTEXH - 1);
    float xf = fminf(fmaxf(floorf(xs), 0.0f), (float)(kTEXW - 2));
    float yf = fminf(fmaxf(floorf(ysv), 0.0f), (float)(kTEXH - 2));
    float fx = xs - xf, fy = ysv - yf;
    int xi = (int)xf, yi = (int)yf;
    int i00 = yi * kTEXW + xi;
    for (int c = 0; c < 3; ++c) {
      const float* tp = tex + (size_t)c * kTEXH * kTEXW;
      float v00 = tp[i00], v01 = tp[i00 + 1];
      float v10 = tp[i00 + kTEXW], v11 = tp[i00 + kTEXW + 1];
      float val = v00 * (1.f - fx) * (1.f - fy) + v01 * fx * (1.f - fy) +
                  v10 * (1.f - fx) * fy + v11 * fx * fy;
      ws[WS_RGB + ((size_t)(r * 3 + c) * kH + y) * kW + px] = val;
    }
    ws[WS_MASK + ((size_t)r * kH + y) * kW + px] = hit;
  }
}

// ---------------- post-processing ----------------
__global__ void erode_kernel(const float* __restrict__ src,
                             float* __restrict__ dst) {
  int idx = blockIdx.x * blockDim.x + threadIdx.x;
  if (idx >= kNR * kHW) return;
  int x = idx & (kW - 1);
  int y = (idx / kW) & (kH - 1);
  int r = idx / kHW;
  float m = INFINITY;
  for (int dy = -1; dy <= 1; ++dy) {
    int yy = y + dy;
    if (yy < 0 || yy >= kH) continue;
    for (int dx = -1; dx <= 1; ++dx) {
      int xx = x + dx;
      if (xx < 0 || xx >= kW) continue;
      m = fminf(m, src[(size_t)r * kHW + yy * kW + xx]);
    }
  }
  dst[idx] = m;
}

__global__ void blurH_kernel(const float* __restrict__ src,
                             float* __restrict__ dst, int nImg) {
  int idx = blockIdx.x * blockDim.x + threadIdx.x;
  if (idx >= nImg * kHW) return;
  int x = idx & (kW - 1);
  int base = idx - x;
  float s = 0.f;
#pragma unroll
  for (int k = 0; k < 11; ++k) {
    int xx = x + k - 5;
    if (xx >= 0 && xx < kW) s += GW[k] * src[base + xx];
  }
  dst[idx] = s;
}

__global__ void blurV_kernel(const float* __restrict__ src,
                             float* __restrict__ dst, int nImg) {
  int idx = blockIdx.x * blockDim.x + threadIdx.x;
  if (idx >= nImg * kHW) return;
  int x = idx & (kW - 1);
  int y = (idx / kW) & (kH - 1);
  int img = idx / kHW;
  float s = 0.f;
#pragma unroll
  for (int k = 0; k < 11; ++k) {
    int yy = y + k - 5;
    if (yy >= 0 && yy < kH) s += GW[k] * src[(size_t)img * kHW + yy * kW + x];
  }
  dst[idx] = s;
}

// vertical blur writing straight into the output tensor layout
// out[((render*4 + chOff + c)*H + y)*W + x]
__global__ void blurV_out_kernel(const float* __restrict__ src,
                                 float* __restrict__ out, int chPer,
                                 int chOff) {
  int idx = blockIdx.x * blockDim.x + threadIdx.x;
  if (idx >= kNR * chPer * kHW) return;
  int x = idx & (kW - 1);
  int y = (idx / kW) & (kH - 1);
  int img = idx / kHW;
  int r = img / chPer;
  int c = img - r * chPer;
  float s = 0.f;
#pragma unroll
  for (int k = 0; k < 11; ++k) {
    int yy = y + k - 5;
    if (yy >= 0 && yy < kH) s += GW[k] * src[(size_t)img * kHW + yy * kW + x];
  }
  out[(((size_t)r * 4 + chOff + c) * kH + y) * kW + x] = s;
}

extern "C" void kernel_launch(void* const* d_in, const int* in_sizes, int n_in,
                              void* d_out, int out_size, void* d_ws,
                              size_t ws_size, hipStream_t stream) {
  (void)in_sizes; (void)n_in; (void)out_size; (void)ws_size;
  const float* vertices = (const float*)d_in[0];
  const float* quaternion = (const float*)d_in[1];
  const float* translation = (const float*)d_in[2];
  // d_in[3] = exp (unused by reference)
  const float* face_features = (const float*)d_in[4];
  const float* texture_maps = (const float*)d_in[5];
  const int* faces = (const int*)d_in[6];
  float* out = (float*)d_out;
  float* ws = (float*)d_ws;

  prep_rt_kernel<<<1, 32, 0, stream>>>(quaternion, translation, ws);
  prep_faces_kernel<<<kNR, 256, 0, stream>>>(vertices, faces, ws);
  raster_kernel<<<dim3(kH, kNR), 256, 0, stream>>>(ws, ws, face_features,
                                                   texture_maps);

  erode_kernel<<<(kNR * kHW) / 256, 256, 0, stream>>>(ws + WS_MASK,
                                                      ws + WS_ME);
  // rgb: blur H then V (separable, order commutes), V-pass writes output ch0..2
  blurH_kernel<<<(kNR * 3 * kHW) / 256, 256, 0, stream>>>(ws + WS_RGB,
                                                          ws + WS_T0, kNR * 3);
  blurV_out_kernel<<<(kNR * 3 * kHW) / 256, 256, 0, stream>>>(ws + WS_T0, out,
                                                              3, 0);
  // mask: erode -> blur -> blur, final V-pass writes output ch3
  blurH_kernel<<<(kNR * kHW) / 256, 256, 0, stream>>>(ws + WS_ME, ws + WS_T1,
                                                      kNR);
  blurV_kernel<<<(kNR * kHW) / 256, 256, 0, stream>>>(ws + WS_T1, ws + WS_ME,
                                                      kNR);
  blurH_kernel<<<(kNR * kHW) / 256, 256, 0, stream>>>(ws + WS_ME, ws + WS_T1,
                                                      kNR);
  blurV_out_kernel<<<(kNR * kHW) / 256, 256, 0, stream>>>(ws + WS_T1, out, 1,
                                                          3);
}